// MeanPooling_84473416778474
// MI455X (gfx1250) — compile-verified
//
#include <hip/hip_runtime.h>
#include <hip/hip_bf16.h>

// MeanPooling as batched GEMM on CDNA5 fp32 WMMA (v_wmma_f32_16x16x4_f32).
// pooled[b,n,h] = sum_s M[b,n,s] * D[b,s,h] * (1/len[b,n])
// B=8, N=512, S=4096, H=1024, all fp32 (full-precision matrix path).

typedef float v2f __attribute__((ext_vector_type(2)));
typedef float v4f __attribute__((ext_vector_type(4)));
typedef float v8f __attribute__((ext_vector_type(8)));

#define BATCH 8
#define NDIM  512
#define SDIM  4096
#define HDIM  1024

#define BLK_N 64      // output rows per workgroup
#define BLK_H 128     // output cols per workgroup
#define BLK_K 32      // K-chunk per LDS buffer (double-buffered)
#define A_STRIDE 36   // 32 + 4 pad: 16B-aligned rows, conflict-free 16-row fragment reads
#define BP_STRIDE 288 // k-pair row: 2*128 + 32 pad; 288 % 64 == 32 -> lane halves on disjoint banks
#define NPAIR (BLK_K / 2)

// Stage one K-chunk: A as plain padded rows, B k-pair interleaved so a WMMA
// B-fragment {B[k][c], B[k+1][c]} (k even) is one aligned ds_load_b64.
__device__ __forceinline__ void stage_chunk(const float* __restrict__ Ag,
                                            const float* __restrict__ Bg,
                                            int ks, float* __restrict__ Al,
                                            float* __restrict__ Bl, int tid)
{
    // A: 64 rows x 32 K = 512 float4, 2 per thread (coalesced b128 loads).
    #pragma unroll
    for (int i = 0; i < 2; ++i) {
        int idx = i * 256 + tid;
        int r   = idx >> 3;
        int c   = (idx & 7) << 2;
        v4f v = *(const v4f*)(Ag + (size_t)r * SDIM + ks + c);
        *(v4f*)(&Al[r * A_STRIDE + c]) = v;
    }
    // B: 16 k-pairs x 64 col-pairs = 1024 items, 4 per thread.
    // Read float2 from rows 2p and 2p+1, write interleaved b128:
    //   {B[2p][c], B[2p+1][c], B[2p][c+1], B[2p+1][c+1]}
    #pragma unroll
    for (int i = 0; i < 4; ++i) {
        int idx = i * 256 + tid;
        int p   = idx >> 6;
        int c   = (idx & 63) << 1;
        v2f lo = *(const v2f*)(Bg + (size_t)(ks + 2 * p)     * HDIM + c);
        v2f hi = *(const v2f*)(Bg + (size_t)(ks + 2 * p + 1) * HDIM + c);
        v4f v; v.x = lo.x; v.y = hi.x; v.z = lo.y; v.w = hi.y;
        *(v4f*)(&Bl[p * BP_STRIDE + 2 * c]) = v;
    }
}

// 8 K-steps of 4, 4 column tiles each: 32 v_wmma_f32_16x16x4_f32 per chunk.
__device__ __forceinline__ void compute_chunk(const float* __restrict__ Al,
                                              const float* __restrict__ Bl,
                                              int wr, int wc, int hi, int ln,
                                              v8f& acc0, v8f& acc1,
                                              v8f& acc2, v8f& acc3)
{
    #pragma unroll
    for (int kk = 0; kk < BLK_K; kk += 4) {
        // A fragment: a.x = A[m][kk+2*hi], a.y = A[m][kk+2*hi+1]  (one b64)
        v2f a = *(const v2f*)(&Al[(wr + ln) * A_STRIDE + kk + 2 * hi]);
        // B fragments: pair row (kk>>1)+hi, column tiles 32 floats apart.
        const float* bp = &Bl[((kk >> 1) + hi) * BP_STRIDE + 2 * (wc + ln)];
        v2f b0 = *(const v2f*)(bp +  0);
        acc0 = __builtin_amdgcn_wmma_f32_16x16x4_f32(false, a, false, b0,
                                                     (short)0, acc0, false, false);
        v2f b1 = *(const v2f*)(bp + 32);
        acc1 = __builtin_amdgcn_wmma_f32_16x16x4_f32(false, a, false, b1,
                                                     (short)0, acc1, false, false);
        v2f b2 = *(const v2f*)(bp + 64);
        acc2 = __builtin_amdgcn_wmma_f32_16x16x4_f32(false, a, false, b2,
                                                     (short)0, acc2, false, false);
        v2f b3 = *(const v2f*)(bp + 96);
        acc3 = __builtin_amdgcn_wmma_f32_16x16x4_f32(false, a, false, b3,
                                                     (short)0, acc3, false, false);
    }
}

__global__ __launch_bounds__(256)
void meanpool_wmma_f32(const float* __restrict__ doc_state,     // [B][S][H]
                       const float* __restrict__ nodes_mapping, // [B][N][S]
                       const float* __restrict__ nodes_len,     // [B][N]
                       float* __restrict__ out)                 // [B][N][H]
{
    __shared__ float Al[2][BLK_N * A_STRIDE];   // 2 x  9,216 B
    __shared__ float Bl[2][NPAIR * BP_STRIDE];  // 2 x 18,432 B
    __shared__ float invLen[BLK_N];             //       256 B  (55.5 KB total)

    const int tid  = threadIdx.x;
    const int lane = tid & 31;       // wave32
    const int wave = tid >> 5;       // 8 waves per workgroup
    const int b    = blockIdx.z;
    const int n0   = blockIdx.y * BLK_N;
    const int h0   = blockIdx.x * BLK_H;

    const float* Ag = nodes_mapping + ((size_t)b * NDIM + n0) * SDIM;
    const float* Bg = doc_state     + (size_t)b * SDIM * HDIM + h0;

    if (tid < BLK_N)
        invLen[tid] = 1.0f / nodes_len[(size_t)b * NDIM + n0 + tid];

    // Wave tile: rows wr..wr+15 of the 64-row block, cols wc..wc+63 of 128.
    const int wr = (wave >> 1) * 16;
    const int wc = (wave & 1) * 64;
    const int hi = lane >> 4;
    const int ln = lane & 15;

    v8f acc0 = {}, acc1 = {}, acc2 = {}, acc3 = {};

    // Software pipeline: one barrier per chunk, stage next while computing cur.
    stage_chunk(Ag, Bg, 0, Al[0], Bl[0], tid);

    for (int ks = 0; ks < SDIM; ks += 2 * BLK_K) {
        __syncthreads();  // buf0 ready; prior compute on buf1 drained
        if (ks + BLK_K < SDIM)
            stage_chunk(Ag, Bg, ks + BLK_K, Al[1], Bl[1], tid);
        compute_chunk(Al[0], Bl[0], wr, wc, hi, ln, acc0, acc1, acc2, acc3);

        __syncthreads();  // buf1 ready; compute on buf0 drained
        if (ks + 2 * BLK_K < SDIM)
            stage_chunk(Ag, Bg, ks + 2 * BLK_K, Al[0], Bl[0], tid);
        compute_chunk(Al[1], Bl[1], wr, wc, hi, ln, acc0, acc1, acc2, acc3);
    }

    // Epilogue: C/D layout — VGPR r holds row (r + 8*hi), lane&15 is the column.
    #pragma unroll
    for (int r = 0; r < 8; ++r) {
        const int   rloc = wr + 8 * hi + r;
        const float sc   = invLen[rloc];
        float* orow = out + ((size_t)b * NDIM + n0 + rloc) * HDIM + h0 + wc + ln;
        orow[ 0] = acc0[r] * sc;
        orow[16] = acc1[r] * sc;
        orow[32] = acc2[r] * sc;
        orow[48] = acc3[r] * sc;
    }
}

extern "C" void kernel_launch(void* const* d_in, const int* in_sizes, int n_in,
                              void* d_out, int out_size, void* d_ws, size_t ws_size,
                              hipStream_t stream) {
    (void)in_sizes; (void)n_in; (void)out_size; (void)d_ws; (void)ws_size;
    const float* doc_state     = (const float*)d_in[0];
    const float* nodes_mapping = (const float*)d_in[1];
    const float* nodes_len     = (const float*)d_in[2];
    float* out = (float*)d_out;

    dim3 grid(HDIM / BLK_H, NDIM / BLK_N, BATCH); // 8 x 8 x 8 = 512 workgroups
    dim3 block(256);
    meanpool_wmma_f32<<<grid, block, 0, stream>>>(doc_state, nodes_mapping,
                                                  nodes_len, out);
}